// FourierBasisTRF_41034117545995
// MI455X (gfx1250) — compile-verified
//
#include <hip/hip_runtime.h>

typedef __attribute__((ext_vector_type(2))) float v2f;
typedef __attribute__((ext_vector_type(4))) float v4f;
typedef __attribute__((ext_vector_type(8))) float v8f;

namespace {
constexpr int   kB            = 8;
constexpr int   kW            = 65;     // t = 0..64
constexpr int   kS            = 4096;
constexpr int   kNOI          = 64;     // nOut * nIn
constexpr int   kNB           = 17;     // coefs last-dim
constexpr int   kNGRP         = 4;      // K = 16 harmonics, 4 groups of 4
constexpr int   kNJ           = 4;      // 64 (o,i) columns = 4 tiles of 16
constexpr int   kTileS        = 16;
constexpr int   kTilesPerWave = 8;
constexpr int   kSplit        = kS / kTileS / kTilesPerWave;   // 32
constexpr int   kWavesPerBlk  = 4;
constexpr int   kTotalWaves   = kB * kW * kSplit;              // 16640
constexpr float kOmega  = 0.07853981633974483f;   // 2*pi/80
constexpr float kNorm   = 0.15811388300841897f;   // 1/sqrt(80/2)
constexpr float kConst0 = 0.11180339887498950f;   // kNorm/sqrt(2)
}

__global__ __launch_bounds__(128) void fourier_basis_wmma(
    const float* __restrict__ aIn, const float* __restrict__ bIn,
    const float* __restrict__ cIn, const float* __restrict__ coefs,
    float* __restrict__ out)
{
  const int lane = threadIdx.x & 31;
  const int hi   = lane >> 4;          // 0: supplies K0/K1, 1: supplies K2/K3
  const int lm   = lane & 15;
  const int gw   = blockIdx.x * kWavesPerBlk + (threadIdx.x >> 5);
  if (gw >= kTotalWaves) return;       // wave-uniform

  const int chunk = gw % kSplit;
  const int bw    = gw / kSplit;
  const int w     = bw % kW;
  const int b     = bw / kW;

  // --- Preload constant B fragments (coefs harmonics, 4x16 tiles, K=16) ---
  // Row k of group g holds coefs[..., 1 + 4g + k]; this lane supplies rows 2*hi, 2*hi+1.
  v2f   Bm[kNJ][kNGRP];
  float dcj[kNJ];                      // const0 * coefs[m*17], added in epilogue
  #pragma unroll
  for (int j = 0; j < kNJ; ++j) {
    const int m = j * 16 + lm;                 // flat (o,i) column
    const float* cj = coefs + m * kNB;
    dcj[j] = kConst0 * cj[0];
    #pragma unroll
    for (int g = 0; g < kNGRP; ++g) {
      const int k0 = 1 + g * 4 + 2 * hi;       // 1..16, always in range
      v2f bf = { cj[k0], cj[k0 + 1] };
      Bm[j][g] = bf;
    }
  }

  const float* aRow = aIn + (size_t)b * kS;
  const float* bRow = bIn + (size_t)b * kS;
  const float* cRow = cIn + (size_t)b * kS;
  const size_t outBW = (size_t)b * kNOI * kW * kS + (size_t)w * kS;

  // Cap unrolling: 2 tiles in flight is enough to overlap loads/sincos with
  // WMMA + stores while staying well under 256 VGPRs (keeps occupancy high).
  #pragma clang loop unroll_count(2)
  for (int t = 0; t < kTilesPerWave; ++t) {
    const int s0 = (chunk * kTilesPerWave + t) * kTileS;
    const int sL = s0 + lm;            // site handled by this lane (dup in lane+16)

    // --- per-lane harmonics: lo lanes need orders 1,3,5,7; hi lanes 2,4,6,8 ---
    const float bv = bRow[sL];
    const float cv = cRow[sL];
    const float om = kOmega * (cv * ((float)w - bv));
    float sw, cw;
    __sincosf(om, &sw, &cw);
    const float s2 = 2.0f * sw * cw;          // sin(2w)
    const float c2 = 1.0f - 2.0f * sw * sw;   // cos(2w)
    // this lane's starting harmonic: order (hi + 1)
    float sk = hi ? s2 : sw;
    float ck = hi ? c2 : cw;

    // --- GEMM tile: [16 sites x 16] x [16 x 64] via f32 WMMA ---
    // g = 0 uses the inline-0 C operand form (dc handled in the epilogue).
    v8f acc[kNJ];
    {
      v2f A = { kNorm * sk, kNorm * ck };
      const v8f zero = { 0.f, 0.f, 0.f, 0.f, 0.f, 0.f, 0.f, 0.f };
      #pragma unroll
      for (int j = 0; j < kNJ; ++j) {
        acc[j] = __builtin_amdgcn_wmma_f32_16x16x4_f32(
            false, A, false, Bm[j][0], (short)0, zero, false, false);
      }
      const float sn = sk * c2 + ck * s2;
      const float cn = ck * c2 - sk * s2;
      sk = sn; ck = cn;
    }
    #pragma unroll
    for (int g = 1; g < kNGRP; ++g) {
      v2f A = { kNorm * sk, kNorm * ck };
      #pragma unroll
      for (int j = 0; j < kNJ; ++j) {
        acc[j] = __builtin_amdgcn_wmma_f32_16x16x4_f32(
            false, A, false, Bm[j][g], (short)0, acc[j], false, false);
      }
      // advance harmonic order by 2 (complex multiply by (c2, s2))
      const float sn = sk * c2 + ck * s2;
      const float cn = ck * c2 - sk * s2;
      sk = sn; ck = cn;
    }

    // --- epilogue: out = a * (dc + gemm); each lane owns 8 consecutive s ---
    const int sA = s0 + hi * 8;
    const v4f a0 = *(const v4f*)(aRow + sA);
    const v4f a1 = *(const v4f*)(aRow + sA + 4);

    #pragma unroll
    for (int j = 0; j < kNJ; ++j) {
      const float dc = dcj[j];
      v8f d = acc[j];
      d[0] = (d[0] + dc) * a0[0]; d[1] = (d[1] + dc) * a0[1];
      d[2] = (d[2] + dc) * a0[2]; d[3] = (d[3] + dc) * a0[3];
      d[4] = (d[4] + dc) * a1[0]; d[5] = (d[5] + dc) * a1[1];
      d[6] = (d[6] + dc) * a1[2]; d[7] = (d[7] + dc) * a1[3];

      const int m = j * 16 + lm;
      float* p = out + outBW + (size_t)m * ((size_t)kW * kS) + sA;
      v4f lo4 = { d[0], d[1], d[2], d[3] };
      v4f hi4 = { d[4], d[5], d[6], d[7] };
      __builtin_nontemporal_store(lo4, (v4f*)p);
      __builtin_nontemporal_store(hi4, (v4f*)(p + 4));
    }
  }
}

extern "C" void kernel_launch(void* const* d_in, const int* in_sizes, int n_in,
                              void* d_out, int out_size, void* d_ws, size_t ws_size,
                              hipStream_t stream) {
  const float* a     = (const float*)d_in[0];
  const float* b     = (const float*)d_in[1];
  const float* c     = (const float*)d_in[2];
  const float* coefs = (const float*)d_in[3];
  float* out = (float*)d_out;

  dim3 grid(kTotalWaves / kWavesPerBlk);   // 4160
  dim3 block(kWavesPerBlk * 32);           // 128
  fourier_basis_wmma<<<grid, block, 0, stream>>>(a, b, c, coefs, out);
}